// SelfAttention_87720412053840
// MI455X (gfx1250) — compile-verified
//
#include <hip/hip_runtime.h>
#include <cstddef>
#include <cstdint>

#define NEG_INF (-1.0e30f)

constexpr int Bn = 32;
constexpr int Ln = 1024;
constexpr int Hn = 512;

typedef __attribute__((ext_vector_type(2))) float v2f;
typedef __attribute__((ext_vector_type(8))) float v8f;

__device__ __forceinline__ v8f wmma_f32(v2f a, v2f b, v8f c) {
    // D = A(16x4) * B(4x16) + C(16x16), fp32, wave32
    return __builtin_amdgcn_wmma_f32_16x16x4_f32(
        false, a, false, b, (short)0, c, false, false);
}

// Async DMA: 16B per lane, global -> LDS, tracked by ASYNCcnt (CDNA5).
__device__ __forceinline__ void async_copy16(uint32_t lds_off, const void* gaddr) {
    asm volatile("global_load_async_to_lds_b128 %0, %1, off"
                 :: "v"(lds_off), "v"(gaddr) : "memory");
}
template <int N>
__device__ __forceinline__ void wait_asynccnt() {
    asm volatile("s_wait_asynccnt %0" :: "i"(N) : "memory");
}
__device__ __forceinline__ uint32_t lds_offset(const void* p) {
    // addrspacecast(local->flat) = aperture_hi:offset; low 32 bits are the
    // wave-relative LDS byte offset expected by async-to-LDS VDST.
    return (uint32_t)(uintptr_t)p;
}

// ---------------------------------------------------------------------------
// K-1: Cq = c * cq_weight (broadcast over H)
// ---------------------------------------------------------------------------
__global__ void k_scale(const float* __restrict__ c, const float* __restrict__ cqw,
                        float* __restrict__ Cq) {
    size_t idx = (size_t)blockIdx.x * 256 + threadIdx.x;  // < B*L*H
    int h = (int)(idx & (Hn - 1));
    Cq[idx] = c[idx] * cqw[h];
}

// ---------------------------------------------------------------------------
// K0: s0[b,l] = c[b,l,:] . c_weight ; s1[b,l] = c[b,l,:] . q_weight
// ---------------------------------------------------------------------------
__global__ void k_rowdots(const float* __restrict__ c,
                          const float* __restrict__ cw,
                          const float* __restrict__ qw,
                          float* __restrict__ s0, float* __restrict__ s1) {
    int wave = threadIdx.x >> 5;
    int lane = threadIdx.x & 31;
    int row  = blockIdx.x * 8 + wave;  // over B*L
    const float* cr = c + (size_t)row * Hn;
    float a0 = 0.f, a1 = 0.f;
    for (int i = 0; i < Hn; i += 32) {
        float v = cr[i + lane];
        a0 += v * cw[i + lane];
        a1 += v * qw[i + lane];
    }
    for (int off = 16; off > 0; off >>= 1) {
        a0 += __shfl_xor(a0, off, 32);
        a1 += __shfl_xor(a1, off, 32);
    }
    if (lane == 0) { s0[row] = a0; s1[row] = a1; }
}

// ---------------------------------------------------------------------------
// K1: S = Cq @ c^T + s0[row] + s1[col] + bias
// 128x64 block tile; 32x32 per wave (4 WMMA accs); K staged 32 at a time via
// async global->LDS DMA, double buffered.
// ---------------------------------------------------------------------------
__global__ void __launch_bounds__(256)
k_gemm_s(const float* __restrict__ Aq, const float* __restrict__ c,
         const float* __restrict__ s0, const float* __restrict__ s1,
         const float* __restrict__ bias, float* __restrict__ S) {
    __shared__ float Al[2][128 * 36];
    __shared__ float Bl[2][64 * 36];

    int bx = blockIdx.x;
    int b  = bx >> 7;
    int rm = bx & 127;
    int i0 = (rm >> 4) * 128;  // 8 row tiles
    int j0 = (rm & 15) * 64;   // 16 col tiles

    const float* Ab = Aq + (size_t)b * Ln * Hn;  // Cq rows
    const float* Bb = c  + (size_t)b * Ln * Hn;  // c rows (B = c^T)

    int t    = threadIdx.x;
    int wave = t >> 5;
    int lane = t & 31;
    int half = lane >> 4;
    int mrow = lane & 15;
    int kadd = half * 2;
    int wm   = wave >> 1;  // 0..3: row offset wm*32
    int wn   = wave & 1;   // col offset wn*32

    uint32_t aBase[2] = { lds_offset(&Al[0][0]), lds_offset(&Al[1][0]) };
    uint32_t bBase[2] = { lds_offset(&Bl[0][0]), lds_offset(&Bl[1][0]) };

    auto issue = [&](int buf, int k0) {
        #pragma unroll
        for (int p = 0; p < 4; ++p) {               // A: 128 rows x 8 chunks
            int idx = (wave * 4 + p) * 32 + lane;   // 0..1023
            int row = idx >> 3, ch = (idx & 7) * 4;
            async_copy16(aBase[buf] + (uint32_t)(row * 36 + ch) * 4,
                         Ab + (size_t)(i0 + row) * Hn + k0 + ch);
        }
        #pragma unroll
        for (int p = 0; p < 2; ++p) {               // B: 64 rows x 8 chunks
            int idx = (wave * 2 + p) * 32 + lane;   // 0..511
            int row = idx >> 3, ch = (idx & 7) * 4;
            async_copy16(bBase[buf] + (uint32_t)(row * 36 + ch) * 4,
                         Bb + (size_t)(j0 + row) * Hn + k0 + ch);
        }
    };

    v8f acc00 = {}, acc01 = {}, acc10 = {}, acc11 = {};

    issue(0, 0);
    constexpr int NST = Hn / 32;  // 16 stages
    for (int s = 0; s < NST; ++s) {
        int cur = s & 1;
        if (s + 1 < NST) { issue(cur ^ 1, (s + 1) * 32); wait_asynccnt<6>(); }
        else             { wait_asynccnt<0>(); }
        __syncthreads();
        const float* A0 = &Al[cur][(wm * 32 + mrow) * 36];
        const float* A1 = A0 + 16 * 36;
        const float* B0 = &Bl[cur][(wn * 32 + mrow) * 36];
        const float* B1 = B0 + 16 * 36;
        #pragma unroll
        for (int kk = 0; kk < 32; kk += 4) {
            int kb = kk + kadd;
            v2f a0 = *(const v2f*)(A0 + kb);
            v2f a1 = *(const v2f*)(A1 + kb);
            v2f b0 = *(const v2f*)(B0 + kb);
            v2f b1 = *(const v2f*)(B1 + kb);
            acc00 = wmma_f32(a0, b0, acc00);
            acc01 = wmma_f32(a0, b1, acc01);
            acc10 = wmma_f32(a1, b0, acc10);
            acc11 = wmma_f32(a1, b1, acc11);
        }
        __syncthreads();
    }

    const float* s0p = s0 + b * Ln;
    const float* s1p = s1 + b * Ln;
    float biasv = bias[0];
    float* Sb = S + (size_t)b * Ln * Ln;
    int colA = j0 + wn * 32 + mrow;
    int colB = colA + 16;
    float sA = s1p[colA], sB = s1p[colB];
    #pragma unroll
    for (int r = 0; r < 8; ++r) {
        int row0 = i0 + wm * 32 + r + half * 8;
        int row1 = row0 + 16;
        float b0v = s0p[row0] + biasv;
        float b1v = s0p[row1] + biasv;
        Sb[(size_t)row0 * Ln + colA] = acc00[r] + b0v + sA;
        Sb[(size_t)row0 * Ln + colB] = acc01[r] + b0v + sB;
        Sb[(size_t)row1 * Ln + colA] = acc10[r] + b1v + sA;
        Sb[(size_t)row1 * Ln + colB] = acc11[r] + b1v + sB;
    }
}

// ---------------------------------------------------------------------------
// K2: column softmax (axis=1), mask on the row index; writes P2T[b,m,l]
// ---------------------------------------------------------------------------
__global__ void __launch_bounds__(256)
k_colsoftmax(const float* __restrict__ S, const int* __restrict__ cmask,
             float* __restrict__ P2T) {
    int b  = blockIdx.x >> 5;
    int m0 = (blockIdx.x & 31) * 32;
    int t = threadIdx.x;
    int cidx = t & 31;
    int ridx = t >> 5;  // 0..7
    int col = m0 + cidx;

    const float* Sb = S + (size_t)b * Ln * Ln;
    const int*   mb = cmask + b * Ln;

    float mx = -3.0e38f, sm = 0.f;
    for (int it = 0; it < 128; ++it) {
        int l = ridx + it * 8;
        float mk = (float)mb[l];
        float v = Sb[(size_t)l * Ln + col];
        v = mk * v + (1.f - mk) * NEG_INF;
        float nm = fmaxf(mx, v);
        sm = sm * __expf(mx - nm) + __expf(v - nm);
        mx = nm;
    }
    __shared__ float shm[8][32];
    __shared__ float shs[8][32];
    shm[ridx][cidx] = mx;
    shs[ridx][cidx] = sm;
    __syncthreads();
    if (ridx == 0) {
        float M = shm[0][cidx], Ss = shs[0][cidx];
        #pragma unroll
        for (int i = 1; i < 8; ++i) {
            float m2 = shm[i][cidx], s2 = shs[i][cidx];
            float nm = fmaxf(M, m2);
            Ss = Ss * __expf(M - nm) + s2 * __expf(m2 - nm);
            M = nm;
        }
        shm[0][cidx] = M;
        shs[0][cidx] = 1.f / Ss;
    }
    __syncthreads();
    float M   = shm[0][cidx];
    float inv = shs[0][cidx];
    float* Pp = P2T + ((size_t)b * Ln + col) * Ln;
    for (int it = 0; it < 128; ++it) {
        int l = ridx + it * 8;
        float mk = (float)mb[l];
        float v = Sb[(size_t)l * Ln + col];
        v = mk * v + (1.f - mk) * NEG_INF;
        Pp[l] = __expf(v - M) * inv;
    }
}

// ---------------------------------------------------------------------------
// K3: row softmax (axis=2), mask on the col index; in place S -> P1
// ---------------------------------------------------------------------------
__global__ void __launch_bounds__(256)
k_rowsoftmax(float* __restrict__ S, const int* __restrict__ cmask) {
    int b = blockIdx.x >> 10;
    float* Srow = S + (size_t)blockIdx.x * Ln;
    const int* mb = cmask + b * Ln;
    int t = threadIdx.x;

    float v[4];
    float mx = -3.0e38f;
    #pragma unroll
    for (int i = 0; i < 4; ++i) {
        int j = t + i * 256;
        float mk = (float)mb[j];
        float x = Srow[j];
        v[i] = mk * x + (1.f - mk) * NEG_INF;
        mx = fmaxf(mx, v[i]);
    }
    __shared__ float red[256];
    red[t] = mx; __syncthreads();
    for (int s = 128; s > 0; s >>= 1) {
        if (t < s) red[t] = fmaxf(red[t], red[t + s]);
        __syncthreads();
    }
    mx = red[0]; __syncthreads();

    float e[4];
    float sm = 0.f;
    #pragma unroll
    for (int i = 0; i < 4; ++i) { e[i] = __expf(v[i] - mx); sm += e[i]; }
    red[t] = sm; __syncthreads();
    for (int s = 128; s > 0; s >>= 1) {
        if (t < s) red[t] += red[t + s];
        __syncthreads();
    }
    float inv = 1.f / red[0];
    #pragma unroll
    for (int i = 0; i < 4; ++i) Srow[t + i * 256] = e[i] * inv;
}

// ---------------------------------------------------------------------------
// K4: C[b] = A[b] (Ln x Ln row-major) @ Bm[b] (Ln x Hn row-major)
// 128x64 tiles, 32x32 per wave, async double-buffered LDS staging.
// ---------------------------------------------------------------------------
__global__ void __launch_bounds__(256)
k_gemm_pn(const float* __restrict__ A, size_t strideA,
          const float* __restrict__ Bm, int ldb, size_t strideB,
          float* __restrict__ C, int ldc, size_t strideC) {
    __shared__ float Al[2][128 * 36];
    __shared__ float Bl[2][32 * 68];

    int bx = blockIdx.x;
    int b  = bx >> 6;
    int rm = bx & 63;
    int i0 = (rm >> 3) * 128;  // 8 row tiles
    int n0 = (rm & 7) * 64;    // 8 col tiles

    const float* Ab = A  + (size_t)b * strideA;
    const float* Bb = Bm + (size_t)b * strideB;
    float*       Cb = C  + (size_t)b * strideC;

    int t    = threadIdx.x;
    int wave = t >> 5;
    int lane = t & 31;
    int half = lane >> 4;
    int mrow = lane & 15;
    int kadd = half * 2;
    int wm   = wave >> 1;
    int wn   = wave & 1;

    uint32_t aBase[2] = { lds_offset(&Al[0][0]), lds_offset(&Al[1][0]) };
    uint32_t bBase[2] = { lds_offset(&Bl[0][0]), lds_offset(&Bl[1][0]) };

    auto issue = [&](int buf, int k0) {
        #pragma unroll
        for (int p = 0; p < 4; ++p) {               // A: 128 rows x 8 chunks
            int idx = (wave * 4 + p) * 32 + lane;   // 0..1023
            int row = idx >> 3, ch = (idx & 7) * 4;
            async_copy16(aBase[buf] + (uint32_t)(row * 36 + ch) * 4,
                         Ab + (size_t)(i0 + row) * Ln + k0 + ch);
        }
        #pragma unroll
        for (int p = 0; p < 2; ++p) {               // B: 32 k-rows x 16 chunks
            int idx = (wave * 2 + p) * 32 + lane;   // 0..511
            int kr = idx >> 4, ch = (idx & 15) * 4;
            async_copy16(bBase[buf] + (uint32_t)(kr * 68 + ch) * 4,
                         Bb + (size_t)(k0 + kr) * ldb + n0 + ch);
        }
    };

    v8f acc00 = {}, acc01 = {}, acc10 = {}, acc11 = {};

    issue(0, 0);
    constexpr int NST = Ln / 32;  // 32 stages
    for (int s = 0; s < NST; ++s) {
        int cur = s & 1;
        if (s + 1 < NST) { issue(cur ^ 1, (s + 1) * 32); wait_asynccnt<6>(); }
        else             { wait_asynccnt<0>(); }
        __syncthreads();
        const float* A0 = &Al[cur][(wm * 32 + mrow) * 36];
        const float* A1 = A0 + 16 * 36;
        const float* Bp = &Bl[cur][0];
        int nA = wn * 32 + mrow;
        int nB = nA + 16;
        #pragma unroll
        for (int kk = 0; kk < 32; kk += 4) {
            int kb = kk + kadd;
            v2f a0 = *(const v2f*)(A0 + kb);
            v2f a1 = *(const v2f*)(A1 + kb);
            v2f b0, b1;
            b0.x = Bp[kb * 68 + nA];
            b0.y = Bp[(kb + 1) * 68 + nA];
            b1.x = Bp[kb * 68 + nB];
            b1.y = Bp[(kb + 1) * 68 + nB];
            acc00 = wmma_f32(a0, b0, acc00);
            acc01 = wmma_f32(a0, b1, acc01);
            acc10 = wmma_f32(a1, b0, acc10);
            acc11 = wmma_f32(a1, b1, acc11);
        }
        __syncthreads();
    }

    int colA = n0 + wn * 32 + mrow;
    int colB = colA + 16;
    #pragma unroll
    for (int r = 0; r < 8; ++r) {
        int row0 = i0 + wm * 32 + r + half * 8;
        int row1 = row0 + 16;
        Cb[(size_t)row0 * ldc + colA] = acc00[r];
        Cb[(size_t)row0 * ldc + colB] = acc01[r];
        Cb[(size_t)row1 * ldc + colA] = acc10[r];
        Cb[(size_t)row1 * ldc + colB] = acc11[r];
    }
}

// ---------------------------------------------------------------------------
// K7: out = [c, a, c*a, c*b]; a already in slice1, raw b in slice3.
// ---------------------------------------------------------------------------
__global__ void k_finish(const float* __restrict__ c, float* __restrict__ out) {
    size_t idx = (size_t)blockIdx.x * 256 + threadIdx.x;  // < B*L*H
    size_t row = idx >> 9;
    int h = (int)(idx & 511);
    float cv = c[idx];
    size_t ob = row * 2048;
    float av = out[ob + 512 + h];
    float bv = out[ob + 1536 + h];
    out[ob + h]        = cv;
    out[ob + 1024 + h] = cv * av;
    out[ob + 1536 + h] = cv * bv;
}

// ---------------------------------------------------------------------------
extern "C" void kernel_launch(void* const* d_in, const int* in_sizes, int n_in,
                              void* d_out, int out_size, void* d_ws, size_t ws_size,
                              hipStream_t stream) {
    const float* c    = (const float*)d_in[0];
    const int*   mask = (const int*)d_in[1];
    const float* cw   = (const float*)d_in[2];
    const float* qw   = (const float*)d_in[3];
    const float* cqw  = (const float*)d_in[4];
    const float* bias = (const float*)d_in[5];
    float* out = (float*)d_out;
    float* ws  = (float*)d_ws;

    float* s0  = ws;                                 // B*L
    float* s1  = s0 + (size_t)Bn * Ln;               // B*L
    float* S   = s1 + (size_t)Bn * Ln;               // B*L*L (becomes P1)
    float* P2T = S + (size_t)Bn * Ln * Ln;           // B*L*L
    float* T   = P2T + (size_t)Bn * Ln * Ln;         // B*L*H
    float* Cq  = T + (size_t)Bn * Ln * Hn;           // B*L*H

    // Cq = c * cq_weight
    k_scale<<<(Bn * Ln * Hn) / 256, 256, 0, stream>>>(c, cqw, Cq);
    // s0, s1
    k_rowdots<<<(Bn * Ln) / 8, 256, 0, stream>>>(c, cw, qw, s0, s1);
    // S = Cq @ c^T + s0 + s1 + bias
    k_gemm_s<<<Bn * 128, 256, 0, stream>>>(Cq, c, s0, s1, bias, S);
    // P2T = softmax over axis=1, transposed
    k_colsoftmax<<<Bn * 32, 256, 0, stream>>>(S, mask, P2T);
    // S -> P1 (row softmax, in place)
    k_rowsoftmax<<<Bn * Ln, 256, 0, stream>>>(S, mask);
    // T = P2T @ c
    k_gemm_pn<<<Bn * 64, 256, 0, stream>>>(P2T, (size_t)Ln * Ln,
                                           c, Hn, (size_t)Ln * Hn,
                                           T, Hn, (size_t)Ln * Hn);
    // a = P1 @ c  -> out slice [H:2H)
    k_gemm_pn<<<Bn * 64, 256, 0, stream>>>(S, (size_t)Ln * Ln,
                                           c, Hn, (size_t)Ln * Hn,
                                           out + Hn, 4 * Hn, (size_t)Ln * 4 * Hn);
    // b_raw = P1 @ T -> out slice [3H:4H)
    k_gemm_pn<<<Bn * 64, 256, 0, stream>>>(S, (size_t)Ln * Ln,
                                           T, Hn, (size_t)Ln * Hn,
                                           out + 3 * Hn, 4 * Hn, (size_t)Ln * 4 * Hn);
    // finish: c, c*a, c*b
    k_finish<<<(Bn * Ln * Hn) / 256, 256, 0, stream>>>(c, out);
}